// Codebook_bk_35639638622554
// MI455X (gfx1250) — compile-verified
//
#include <hip/hip_runtime.h>
#include <hip/hip_bf16.h>
#include <math.h>

typedef __attribute__((ext_vector_type(2))) float v2f;
typedef __attribute__((ext_vector_type(8))) float v8f;

#define NB      16384      // batch rows
#define NE      16384      // codebook entries
#define DF      64         // folded K dimension
#define DFULL   128        // original embedding dim
#define NOUT_ZQ 2097152    // 16384*4*16*2

// ---------------- workspace layout (bytes) ----------------
// acc      @ 0       : 8 doubles  (Szq, Sz, Szq2, Sz2, Szqz, pad)
// colAbs   @ 64      : 128 floats
// normE    @ 1024    : 16384 floats
// idxbuf   @ 66560   : 16384 ints
// a_m2     @ 132096  : 16384*64 floats  (= -2 * zf64, row major)
// bfold    @ 4326400 : 16384*64 floats  (folded codebook, row major)
// total ~ 8.2 MB

__global__ __launch_bounds__(128) void init_kernel(double* acc, float* colAbs) {
    int t = threadIdx.x;
    if (t < 8) acc[t] = 0.0;
    colAbs[t] = 0.0f;
}

// a_m2[b*64 + u] = -(z[b,c,H,2W] + z[b,c,H,2W+1]),  u = c*4 + H*2 + W
__global__ __launch_bounds__(256) void prepa_kernel(const float* __restrict__ z,
                                                    float* __restrict__ a_m2) {
    int i = blockIdx.x * 256 + threadIdx.x;           // 0 .. 16384*64-1
    int b = i >> 6, u = i & 63;
    int c = u >> 2, H = (u >> 1) & 1, W = u & 1;
    const float* zp = z + (size_t)b * DFULL + c * 8 + H * 4 + 2 * W;
    a_m2[i] = -(zp[0] + zp[1]);
}

// bfold[j*64+u] = e[j, c*8 + s] + e[j, c*8 + s + 4]   (u = c*4 + s)
// normE[j]     = sum_k e[j,k]^2  (full 128 dims)
__global__ __launch_bounds__(64) void prepb_kernel(const float* __restrict__ emb,
                                                   float* __restrict__ bfold,
                                                   float* __restrict__ normE) {
    int j = blockIdx.x, u = threadIdx.x;              // u: 0..63
    int c = u >> 2, s = u & 3;
    float e0 = emb[(size_t)j * DFULL + c * 8 + s];
    float e1 = emb[(size_t)j * DFULL + c * 8 + s + 4];
    bfold[(size_t)j * DF + u] = e0 + e1;
    float sq = e0 * e0 + e1 * e1;
    for (int m = 16; m >= 1; m >>= 1) sq += __shfl_xor(sq, m, 32);
    __shared__ float sh[2];
    if ((u & 31) == 0) sh[u >> 5] = sq;
    __syncthreads();
    if (u == 0) normE[j] = sh[0] + sh[1];
}

// colAbs[d] += sum over 256 rows of |emb[r,d]|
__global__ __launch_bounds__(128) void colabs_kernel(const float* __restrict__ emb,
                                                     float* __restrict__ colAbs) {
    int d = threadIdx.x;
    int r0 = blockIdx.x * 256;
    float s = 0.0f;
    for (int r = 0; r < 256; ++r) s += fabsf(emb[(size_t)(r0 + r) * DFULL + d]);
    atomicAdd(&colAbs[d], s);
}

// One workgroup (8 waves) per 32-row block: each wave keeps TWO register-resident
// A tiles (rows +0..15 and +16..31) and runs two independent WMMA chains per
// loaded B tile -> 2x L2 reuse, 2x matrix-pipe ILP.
// score = normE[col] + sum_u a_m2[row,u]*bfold[col,u]   (||e||^2 seeded into C).
__global__ __launch_bounds__(256) void argmin_kernel(const float* __restrict__ a_m2,
                                                     const float* __restrict__ bfold,
                                                     const float* __restrict__ normE,
                                                     int* __restrict__ idxbuf,
                                                     float* __restrict__ idx_out) {
    const int row0 = blockIdx.x * 32;
    const int tid  = threadIdx.x;
    const int wv   = tid >> 5;
    const int lane = tid & 31;
    const int l16  = lane & 15;
    const int hi   = lane >> 4;   // 0: K%4 in {0,1}; 1: K%4 in {2,3}

    // Two A tiles resident for the whole sweep: 16 chunks x float2 per lane each
    v2f a0[16], a1[16];
    const float* abase0 = a_m2 + (size_t)(row0 + l16)      * DF + hi * 2;
    const float* abase1 = a_m2 + (size_t)(row0 + 16 + l16) * DF + hi * 2;
    #pragma unroll
    for (int kc = 0; kc < 16; ++kc) {
        a0[kc] = *(const v2f*)(abase0 + kc * 4);
        a1[kc] = *(const v2f*)(abase1 + kc * 4);
    }

    float best0[8], best1[8]; int bidx0[8], bidx1[8];
    #pragma unroll
    for (int v = 0; v < 8; ++v) {
        best0[v] = 3.4e38f; bidx0[v] = 0;
        best1[v] = 3.4e38f; bidx1[v] = 0;
    }

    for (int ct = wv; ct < NE / 16; ct += 8) {
        const int col0 = ct * 16;
        const float* bbase = bfold + (size_t)(col0 + l16) * DF + hi * 2;
        v2f bb[16];
        #pragma unroll
        for (int kc = 0; kc < 16; ++kc) bb[kc] = *(const v2f*)(bbase + kc * 4);
        float nE = normE[col0 + l16];
        v8f c0 = { nE, nE, nE, nE, nE, nE, nE, nE };
        v8f c1 = c0;
        #pragma unroll
        for (int kc = 0; kc < 16; ++kc) {   // two independent chains share each B
            c0 = __builtin_amdgcn_wmma_f32_16x16x4_f32(
                     false, a0[kc], false, bb[kc], (short)0, c0, false, false);
            c1 = __builtin_amdgcn_wmma_f32_16x16x4_f32(
                     false, a1[kc], false, bb[kc], (short)0, c1, false, false);
        }
        const int colIdx = col0 + l16;
        #pragma unroll
        for (int v = 0; v < 8; ++v) {
            float s0 = c0[v];
            if (s0 < best0[v]) { best0[v] = s0; bidx0[v] = colIdx; }
            float s1 = c1[v];
            if (s1 < best1[v]) { best1[v] = s1; bidx1[v] = colIdx; }
        }
    }

    // reduce across the 16 lanes of each half (lanes share row set, disjoint cols)
    #pragma unroll
    for (int v = 0; v < 8; ++v) {
        float s = best0[v]; int i = bidx0[v];
        #pragma unroll
        for (int m = 8; m >= 1; m >>= 1) {
            float s2 = __shfl_xor(s, m, 32);
            int   i2 = __shfl_xor(i, m, 32);
            if (s2 < s || (s2 == s && i2 < i)) { s = s2; i = i2; }
        }
        best0[v] = s; bidx0[v] = i;

        s = best1[v]; i = bidx1[v];
        #pragma unroll
        for (int m = 8; m >= 1; m >>= 1) {
            float s2 = __shfl_xor(s, m, 32);
            int   i2 = __shfl_xor(i, m, 32);
            if (s2 < s || (s2 == s && i2 < i)) { s = s2; i = i2; }
        }
        best1[v] = s; bidx1[v] = i;
    }

    __shared__ float lbest[8 * 32];
    __shared__ int   lidx [8 * 32];
    if (l16 == 0) {   // lane 0 holds rows r0..r0+7, lane 16 rows r0+8..r0+15
        #pragma unroll
        for (int v = 0; v < 8; ++v) {
            int r = hi * 8 + v;
            lbest[wv * 32 + r]      = best0[v];
            lidx [wv * 32 + r]      = bidx0[v];
            lbest[wv * 32 + 16 + r] = best1[v];
            lidx [wv * 32 + 16 + r] = bidx1[v];
        }
    }
    __syncthreads();
    if (tid < 32) {
        float s = lbest[tid]; int i = lidx[tid];
        for (int w = 1; w < 8; ++w) {
            float s2 = lbest[w * 32 + tid]; int i2 = lidx[w * 32 + tid];
            if (s2 < s || (s2 == s && i2 < i)) { s = s2; i = i2; }
        }
        idxbuf [row0 + tid] = i;
        idx_out[row0 + tid] = (float)i;
    }
}

// z_q gather, transposed store out[b,w,c,h], and double-precision moments
__global__ __launch_bounds__(128) void gather_kernel(const float* __restrict__ z,
                                                     const float* __restrict__ emb,
                                                     const int* __restrict__ idxbuf,
                                                     float* __restrict__ zq_out,
                                                     double* __restrict__ acc) {
    int b = blockIdx.x, t = threadIdx.x;     // t = c*8 + h*4 + w
    int id = idxbuf[b];
    float e  = emb[(size_t)id * DFULL + t];
    float zv = z  [(size_t)b  * DFULL + t];
    int c = t >> 3, h = (t >> 2) & 1, w = t & 3;
    zq_out[(size_t)b * DFULL + w * 32 + c * 2 + h] = e;

    float vals[5] = { e, zv, e * e, zv * zv, e * zv };
    __shared__ double sdata[5][4];
    int lane = t & 31, wv = t >> 5;
    #pragma unroll
    for (int q = 0; q < 5; ++q) {
        float s = vals[q];
        for (int m = 16; m >= 1; m >>= 1) s += __shfl_xor(s, m, 32);
        if (lane == 0) sdata[q][wv] = (double)s;
    }
    __syncthreads();
    if (t < 5) {
        double s = sdata[t][0] + sdata[t][1] + sdata[t][2] + sdata[t][3];
        atomicAdd(&acc[t], s);
    }
}

__global__ __launch_bounds__(128) void finalize_kernel(const double* __restrict__ acc,
                                                       const float* __restrict__ colAbs,
                                                       float* __restrict__ loss_out) {
    int t = threadIdx.x;
    float m = colAbs[t];
    for (int msk = 16; msk >= 1; msk >>= 1) m = fmaxf(m, __shfl_xor(m, msk, 32));
    __shared__ float sh[4];
    if ((t & 31) == 0) sh[t >> 5] = m;
    __syncthreads();
    if (t == 0) {
        float mx = fmaxf(fmaxf(sh[0], sh[1]), fmaxf(sh[2], sh[3]));
        double Szq = acc[0], Sz = acc[1], Szq2 = acc[2], Sz2 = acc[3], Szqz = acc[4];
        const double N = (double)NB * (double)DFULL;
        double commit  = 1.25 * (Szq2 - 2.0 * Szqz + Sz2) / N;   // (1+BETA)*mean((zq-z)^2)
        double mux = Szq / N, muy = Sz / N;
        double cov = Szqz - N * mux * muy;
        double vx  = Szq2 - N * mux * mux;
        double vy  = Sz2  - N * muy * muy;
        double pearson = 0.5 + 0.5 * cov / (sqrt(vx) * sqrt(vy));
        double loss = commit + pearson + 0.01 * (double)mx;
        *loss_out = (float)loss;
    }
}

extern "C" void kernel_launch(void* const* d_in, const int* in_sizes, int n_in,
                              void* d_out, int out_size, void* d_ws, size_t ws_size,
                              hipStream_t stream) {
    const float* z   = (const float*)d_in[0];   // (16384,16,2,4)
    const float* emb = (const float*)d_in[1];   // (16384,128)

    float* out      = (float*)d_out;
    float* zq_out   = out;                 // 2,097,152 floats
    float* loss_out = out + NOUT_ZQ;       // 1 float
    float* idx_out  = out + NOUT_ZQ + 1;   // 16384 floats

    char*   ws     = (char*)d_ws;
    double* acc    = (double*)(ws + 0);
    float*  colAbs = (float*)(ws + 64);
    float*  normE  = (float*)(ws + 1024);
    int*    idxbuf = (int*)(ws + 66560);
    float*  a_m2   = (float*)(ws + 132096);
    float*  bfold  = (float*)(ws + 4326400);

    init_kernel  <<<1,     128, 0, stream>>>(acc, colAbs);
    prepa_kernel <<<4096,  256, 0, stream>>>(z, a_m2);
    prepb_kernel <<<NE,     64, 0, stream>>>(emb, bfold, normE);
    colabs_kernel<<<64,    128, 0, stream>>>(emb, colAbs);
    argmin_kernel<<<NB/32, 256, 0, stream>>>(a_m2, bfold, normE, idxbuf, idx_out);
    gather_kernel<<<NB,    128, 0, stream>>>(z, emb, idxbuf, zq_out, acc);
    finalize_kernel<<<1,   128, 0, stream>>>(acc, colAbs, loss_out);
}